// DGCFModel_47888885350521
// MI455X (gfx1250) — compile-verified
//
#include <hip/hip_runtime.h>
#include <hip/hip_bf16.h>
#include <stdint.h>

// ---------------- problem constants (match reference) ----------------
#define NUM_USERS 30000
#define NUM_ITEMS 41000
#define NUM_NODE  71000            // NUM_USERS + NUM_ITEMS
#define EMBED_K   64
#define INTENTS   4
#define CHUNK     16               // EMBED_K / INTENTS
#define ROUTING_ITERS 2

#define TPB 256                    // 8 wave32 per block; == EPB*INTENTS
#define EPB 64                     // edges staged per block in scatter kernel

// ---------------- CDNA5 async global->LDS helpers ----------------
typedef int __attribute__((address_space(1))) as1_int;
typedef int __attribute__((address_space(3))) as3_int;

#if defined(__HIP_DEVICE_COMPILE__) && __has_builtin(__builtin_amdgcn_global_load_async_to_lds_b32)
#define HAVE_ASYNC_LDS 1
#endif

__device__ __forceinline__ void stage_b32(const void* g, void* l) {
#if defined(HAVE_ASYNC_LDS)
    // gfx1250: GLOBAL_LOAD_ASYNC_TO_LDS_B32, tracked by ASYNCcnt
    __builtin_amdgcn_global_load_async_to_lds_b32((as1_int*)g, (as3_int*)l, 0, 0);
#else
    *(uint32_t*)l = *(const uint32_t*)g;
#endif
}

__device__ __forceinline__ void wait_async_lds() {
#if defined(HAVE_ASYNC_LDS)
#if __has_builtin(__builtin_amdgcn_s_wait_asynccnt)
    __builtin_amdgcn_s_wait_asynccnt(0);
#else
    asm volatile("s_wait_asynccnt 0x0" ::: "memory");  // guarantee ASYNCcnt==0 before barrier
#endif
#endif
}

// ---------------- small helpers ----------------
__device__ __forceinline__ float4 softmax4(float4 a) {
    float m  = fmaxf(fmaxf(a.x, a.y), fmaxf(a.z, a.w));
    float e0 = expf(a.x - m), e1 = expf(a.y - m), e2 = expf(a.z - m), e3 = expf(a.w - m);
    float inv = 1.0f / (e0 + e1 + e2 + e3);
    return make_float4(e0 * inv, e1 * inv, e2 * inv, e3 * inv);
}

// ---------------- kernels ----------------
// ego = concat(Gu, Gi), float4-vectorized (EMBED_K=64 -> 16 float4 per row)
__global__ void __launch_bounds__(TPB)
k_init_ego4(const float4* __restrict__ Gu4, const float4* __restrict__ Gi4,
            float4* __restrict__ ego4, size_t n4) {
    size_t i = (size_t)blockIdx.x * TPB + threadIdx.x;
    if (i >= n4) return;
    size_t node = i >> 4;
    ego4[i] = (node < NUM_USERS) ? Gu4[i] : Gi4[i - (size_t)NUM_USERS * (EMBED_K / 4)];
}

__global__ void __launch_bounds__(TPB)
k_fill4(float4* __restrict__ p, float v, size_t n4) {
    size_t i = (size_t)blockIdx.x * TPB + threadIdx.x;
    if (i < n4) p[i] = make_float4(v, v, v, v);
}

// per edge: softmax(A[e]) and atomic-accumulate per-intent degree at head node
__global__ void __launch_bounds__(TPB)
k_degree(const int* __restrict__ heads, const float* __restrict__ A,
         float* __restrict__ d, int E) {
    int e = blockIdx.x * TPB + threadIdx.x;
    if (e >= E) return;
    int h = heads[e];
    float4 na = softmax4(((const float4*)A)[e]);
    atomicAdd(&d[h * INTENTS + 0], na.x);
    atomicAdd(&d[h * INTENTS + 1], na.y);
    atomicAdd(&d[h * INTENTS + 2], na.z);
    atomicAdd(&d[h * INTENTS + 3], na.w);
}

__global__ void __launch_bounds__(TPB)
k_rsqrt(float* __restrict__ d, int n) {
    int i = blockIdx.x * TPB + threadIdx.x;
    if (i < n) d[i] = rsqrtf(fmaxf(d[i], 1e-8f));   // 1/sqrt(clip(d,1e-8))
}

// message passing: factor[head] += ego[tail] * (softmax(A)[q] * d[head,q] * d[tail,q])
// Phase 1: async-stage EPB edges' indices + logits into LDS (CDNA5 ASYNCcnt path).
// Phase 2: one thread per (edge,intent) computes the fused edge weight once.
// Phase 3: 64 threads per edge (2 wave32 -> two coalesced 128B half-rows) do the
//          gather + L2-resident f32 atomic scatter; all VALU already hoisted.
__global__ void __launch_bounds__(TPB)
k_scatter(const int* __restrict__ heads, const int* __restrict__ tails,
          const float* __restrict__ A, const float* __restrict__ dinv,
          const float* __restrict__ ego, float* __restrict__ factor, int E) {
    __shared__ int   sh[EPB];
    __shared__ int   st[EPB];
    __shared__ __align__(16) float sA[EPB * INTENTS];
    __shared__ float sW[EPB * INTENTS];

    const int tid  = threadIdx.x;
    const int base = blockIdx.x * EPB;
    const int nE   = min(EPB, E - base);
    if (nE <= 0) return;

    // ---- phase 1: stage EPB heads + EPB tails + EPB*4 logits (384 dwords) async
    for (int i = tid; i < EPB * 6; i += TPB) {
        if (i < EPB) {
            if (i < nE) stage_b32(heads + base + i, &sh[i]);
        } else if (i < 2 * EPB) {
            int j = i - EPB;
            if (j < nE) stage_b32(tails + base + j, &st[j]);
        } else {
            int j = i - 2 * EPB;
            if ((j >> 2) < nE) stage_b32(A + (size_t)base * INTENTS + j, &sA[j]);
        }
    }
    // speculative prefetch of the next tile's indices into L2
    if (tid < EPB) {
        int nb = base + EPB + tid;
        if (nb < E) {
            __builtin_prefetch(heads + nb, 0, 1);
            __builtin_prefetch(tails + nb, 0, 1);
        }
    }
    wait_async_lds();
    __syncthreads();

    // ---- phase 2: fused weight per (edge,intent); TPB == EPB*INTENTS
    {
        const int eo = tid >> 2;
        const int q  = tid & 3;
        if (eo < nE) {
            float4 na = softmax4(*(const float4*)&sA[eo * INTENTS]);
            float naq = (q == 0) ? na.x : (q == 1) ? na.y : (q == 2) ? na.z : na.w;
            int h = sh[eo], t = st[eo];
            sW[tid] = naq * dinv[h * INTENTS + q] * dinv[t * INTENTS + q];
        }
    }
    __syncthreads();

    // ---- phase 3: gather + atomic scatter
    const int c = tid & (EMBED_K - 1);   // column 0..63 in embedding row
    const int q = c >> 4;                // intent factor
    for (int eo = (tid >> 6); eo < nE; eo += TPB / EMBED_K) {
        const int h = sh[eo];
        const int t = st[eo];
        float v = ego[(size_t)t * EMBED_K + c] * sW[eo * INTENTS + q];
        atomicAdd(&factor[(size_t)h * EMBED_K + c], v);
    }
}

// routing update: A[e,q] += <l2norm(factor[head,q,:]), tanh(l2norm(ego[tail,q,:]))>
// one thread per (edge,intent); 16-wide chunk handled via four float4 loads.
__global__ void __launch_bounds__(TPB)
k_routing(const int* __restrict__ heads, const int* __restrict__ tails,
          const float* __restrict__ factor, const float* __restrict__ ego,
          float* __restrict__ A, int E) {
    int idx = blockIdx.x * TPB + threadIdx.x;
    if (idx >= E * INTENTS) return;
    int e = idx >> 2;
    int q = idx & 3;
    int h = heads[e];
    int t = tails[e];
    const float4* hp = (const float4*)(factor + (size_t)h * EMBED_K + q * CHUNK);
    const float4* tp = (const float4*)(ego    + (size_t)t * EMBED_K + q * CHUNK);
    float4 h0 = hp[0], h1 = hp[1], h2 = hp[2], h3 = hp[3];
    float4 t0 = tp[0], t1 = tp[1], t2 = tp[2], t3 = tp[3];

    auto d4 = [] __device__ (float4 a, float4 b) {
        return a.x * b.x + a.y * b.y + a.z * b.z + a.w * b.w;
    };
    float nh = sqrtf(d4(h0, h0) + d4(h1, h1) + d4(h2, h2) + d4(h3, h3));
    float nt = sqrtf(d4(t0, t0) + d4(t1, t1) + d4(t2, t2) + d4(t3, t3));
    float ih = 1.0f / fmaxf(nh, 1e-12f);
    float it = 1.0f / fmaxf(nt, 1e-12f);

    auto acc = [&] __device__ (float4 a, float4 b) {
        return (a.x * ih) * tanhf(b.x * it) + (a.y * ih) * tanhf(b.y * it) +
               (a.z * ih) * tanhf(b.z * it) + (a.w * ih) * tanhf(b.w * it);
    };
    float s = acc(h0, t0) + acc(h1, t1) + acc(h2, t2) + acc(h3, t3);
    A[idx] += s;   // unique (e,q) per thread -> no atomic needed
}

__global__ void __launch_bounds__(TPB)
k_final4(const float4* __restrict__ ego4, const float4* __restrict__ factor4,
         float4* __restrict__ out4, size_t n4) {
    size_t i = (size_t)blockIdx.x * TPB + threadIdx.x;
    if (i >= n4) return;
    float4 a = ego4[i], b = factor4[i];
    out4[i] = make_float4(0.5f * (a.x + b.x), 0.5f * (a.y + b.y),
                          0.5f * (a.z + b.z), 0.5f * (a.w + b.w));
}

// ---------------- host launcher ----------------
extern "C" void kernel_launch(void* const* d_in, const int* in_sizes, int n_in,
                              void* d_out, int out_size, void* d_ws, size_t ws_size,
                              hipStream_t stream) {
    (void)n_in; (void)out_size; (void)ws_size;
    const float* Gu    = (const float*)d_in[0];
    const float* Gi    = (const float*)d_in[1];
    const int*   heads = (const int*)d_in[2];
    const int*   tails = (const int*)d_in[3];
    const int    E     = in_sizes[2];          // 2 * E_HALF directed edges
    float*       out   = (float*)d_out;

    // workspace layout (floats): ego[N*K] | factor[N*K] | d[N*4] | A[E*4]  (~70 MB)
    const size_t NK = (size_t)NUM_NODE * EMBED_K;
    const size_t N4 = (size_t)NUM_NODE * INTENTS;
    const size_t EA = (size_t)E * INTENTS;
    float* ego    = (float*)d_ws;
    float* factor = ego + NK;
    float* dinv   = factor + NK;
    float* Aval   = dinv + N4;

    const int gNK4  = (int)((NK / 4 + TPB - 1) / TPB);
    const int gN44  = (int)((N4 / 4 + TPB - 1) / TPB);
    const int gEA4  = (int)((EA / 4 + TPB - 1) / TPB);
    const int gE    = (E + TPB - 1) / TPB;
    const int gEA   = (int)((EA + TPB - 1) / TPB);
    const int gN4   = (int)((N4 + TPB - 1) / TPB);
    const int gScat = (E + EPB - 1) / EPB;

    k_init_ego4<<<gNK4, TPB, 0, stream>>>((const float4*)Gu, (const float4*)Gi,
                                          (float4*)ego, NK / 4);
    k_fill4<<<gEA4, TPB, 0, stream>>>((float4*)Aval, 1.0f, EA / 4);

    for (int t = 0; t < ROUTING_ITERS; ++t) {
        k_fill4<<<gN44, TPB, 0, stream>>>((float4*)dinv, 0.0f, N4 / 4);
        k_degree<<<gE, TPB, 0, stream>>>(heads, Aval, dinv, E);
        k_rsqrt<<<gN4, TPB, 0, stream>>>(dinv, (int)N4);
        k_fill4<<<gNK4, TPB, 0, stream>>>((float4*)factor, 0.0f, NK / 4);
        k_scatter<<<gScat, TPB, 0, stream>>>(heads, tails, Aval, dinv, ego, factor, E);
        if (t < ROUTING_ITERS - 1)
            k_routing<<<gEA, TPB, 0, stream>>>(heads, tails, factor, ego, Aval, E);
    }
    k_final4<<<gNK4, TPB, 0, stream>>>((const float4*)ego, (const float4*)factor,
                                       (float4*)out, NK / 4);
}